// HC_PRE_77395310674150
// MI455X (gfx1250) — compile-verified
//
#include <hip/hip_runtime.h>

#define E 128            // NUM_EXPERTS (reference constant; kernel layout assumes 128)
#define MAX_ITERS 8      // SINKHORN_ITERS (reference constant)

typedef float v2f __attribute__((ext_vector_type(2)));
typedef float v4f __attribute__((ext_vector_type(4)));
typedef float v8f __attribute__((ext_vector_type(8)));

// Full-wave (32-lane) sum of 2 values per lane, via V_WMMA_F32_16X16X4_F32.
// B = all-ones, C = 0  =>  D[m,j] = sum_k A[m,k].  Summing all 8 D VGPRs per
// lane and adding the shfl_xor(16) partner yields sum over ALL 64 A elements,
// i.e. the wave-wide sum of (p0+p1) — independent of the exact A/B VGPR layout.
__device__ __forceinline__ float wave_sum_wmma(float p0, float p1) {
    v2f a; a[0] = p0; a[1] = p1;
    v2f b; b[0] = 1.0f; b[1] = 1.0f;
    v8f c = {};
    c = __builtin_amdgcn_wmma_f32_16x16x4_f32(
        /*neg_a=*/false, a, /*neg_b=*/false, b,
        /*c_mod=*/(short)0, c, /*reuse_a=*/false, /*reuse_b=*/false);
    float t = ((c[0] + c[1]) + (c[2] + c[3])) + ((c[4] + c[5]) + (c[6] + c[7]));
    t += __shfl_xor(t, 16, 32);
    return t;  // all 32 lanes hold the full sum
}

// ---------------------------------------------------------------------------
// Kernel 0: init small state: d1 = 1, acc = 0
__global__ void hc_init_kernel(float* __restrict__ d1, float* __restrict__ acc) {
    int t = threadIdx.x;
    if (t < E) { d1[t] = 1.0f; acc[t] = 0.0f; }
}

// ---------------------------------------------------------------------------
// Kernel 1: cost = activation(a*x + b); also d0 = 1/T (iters==0 fallback).
// One wave per row: lane l owns columns 4l..4l+3 (coalesced 128-bit loads).
__global__ void hc_act_kernel(const float* __restrict__ x,
                              float* __restrict__ cost,
                              float* __restrict__ d0,
                              const int* __restrict__ hc_fn,
                              const float* __restrict__ hc_scale,
                              const float* __restrict__ hc_base,
                              const float* __restrict__ hc_mult,
                              int T) {
    const int lane = threadIdx.x & 31;
    const int wid  = (blockIdx.x * blockDim.x + threadIdx.x) >> 5;
    const int nw   = (gridDim.x * blockDim.x) >> 5;
    const float a  = hc_mult[0] * hc_scale[0];
    const float b  = hc_base[0];
    const int  fn  = hc_fn[0];
    const float invT = 1.0f / (float)T;

    for (int row = wid; row < T; row += nw) {
        const v4f* xr = (const v4f*)(x + (size_t)row * E);
        v4f xv = __builtin_nontemporal_load(&xr[lane]);   // x is read-once: NT
        v4f l = xv * a + b;
        v4f cv;
        if (fn == 1) {            // sigmoid
            cv.x = 1.0f / (1.0f + expf(-l.x));
            cv.y = 1.0f / (1.0f + expf(-l.y));
            cv.z = 1.0f / (1.0f + expf(-l.z));
            cv.w = 1.0f / (1.0f + expf(-l.w));
        } else if (fn == 2) {     // softmax over the row
            float m = fmaxf(fmaxf(l.x, l.y), fmaxf(l.z, l.w));
            for (int o = 16; o >= 1; o >>= 1) m = fmaxf(m, __shfl_xor(m, o, 32));
            cv.x = expf(l.x - m); cv.y = expf(l.y - m);
            cv.z = expf(l.z - m); cv.w = expf(l.w - m);
            float s = (cv.x + cv.y) + (cv.z + cv.w);
            for (int o = 16; o >= 1; o >>= 1) s += __shfl_xor(s, o, 32);
            float r = 1.0f / s;
            cv = cv * r;
        } else {                  // exp (canonical Sinkhorn)
            cv.x = expf(l.x); cv.y = expf(l.y);
            cv.z = expf(l.z); cv.w = expf(l.w);
        }
        ((v4f*)(cost + (size_t)row * E))[lane] = cv;  // RT store: keep resident in L2
        if (lane == 0) d0[row] = invT;
    }
}

// ---------------------------------------------------------------------------
// Kernel 2 (x8): one fused Sinkhorn iteration, unrolled 2 rows / iteration.
//   s      = cost[t,:] . d1          (WMMA wave reduction)
//   d0[t]  = invT / (s + eps)        (v_rcp_f32 fast path)
//   acc[e] += sum_t d0[t]*cost[t,e]  (per-lane regs -> LDS -> global atomics)
__global__ void hc_pass_kernel(const float* __restrict__ cost,
                               float* __restrict__ d0,
                               const float* __restrict__ d1,
                               float* __restrict__ acc,
                               const float* __restrict__ hc_eps,
                               const int* __restrict__ itersp,
                               int iter, int T) {
    if (iter >= itersp[0]) return;   // uniform across the whole grid
    __shared__ float colacc[E];
    const int tid  = threadIdx.x;
    const int lane = tid & 31;
    for (int i = tid; i < E; i += blockDim.x) colacc[i] = 0.0f;
    __syncthreads();

    const int wid = (blockIdx.x * blockDim.x + tid) >> 5;
    const int nw  = (gridDim.x * blockDim.x) >> 5;
    const float eps  = hc_eps[0];
    const float invT = 1.0f / (float)T;
    const v4f w = ((const v4f*)d1)[lane];   // d1[4*lane .. 4*lane+3]

    v4f lacc = {0.f, 0.f, 0.f, 0.f};
    int row = wid;
    for (; row + nw < T; row += 2 * nw) {   // two independent b128 loads in flight
        const v4f cv0 = ((const v4f*)(cost + (size_t)row * E))[lane];
        const v4f cv1 = ((const v4f*)(cost + (size_t)(row + nw) * E))[lane];
        float s0 = wave_sum_wmma(cv0.x * w.x + cv0.y * w.y,
                                 cv0.z * w.z + cv0.w * w.w);
        float s1 = wave_sum_wmma(cv1.x * w.x + cv1.y * w.y,
                                 cv1.z * w.z + cv1.w * w.w);
        float dn0 = invT * __builtin_amdgcn_rcpf(s0 + eps);
        float dn1 = invT * __builtin_amdgcn_rcpf(s1 + eps);
        if (lane < 2) d0[row + lane * nw] = (lane == 0) ? dn0 : dn1;
        lacc = lacc + cv0 * dn0 + cv1 * dn1;
    }
    for (; row < T; row += nw) {            // tail (not taken when T % (2*nw) == 0)
        const v4f cv = ((const v4f*)(cost + (size_t)row * E))[lane];
        float s = wave_sum_wmma(cv.x * w.x + cv.y * w.y,
                                cv.z * w.z + cv.w * w.w);
        float dn = invT * __builtin_amdgcn_rcpf(s + eps);
        if (lane == 0) d0[row] = dn;
        lacc = lacc + cv * dn;
    }
    atomicAdd(&colacc[4 * lane + 0], lacc.x);
    atomicAdd(&colacc[4 * lane + 1], lacc.y);
    atomicAdd(&colacc[4 * lane + 2], lacc.z);
    atomicAdd(&colacc[4 * lane + 3], lacc.w);
    __syncthreads();
    for (int i = tid; i < E; i += blockDim.x) atomicAdd(&acc[i], colacc[i]);
}

// ---------------------------------------------------------------------------
// Kernel 3 (x8): finalize d1 for this iteration, reset acc for the next.
__global__ void hc_fin_kernel(float* __restrict__ d1, float* __restrict__ acc,
                              const float* __restrict__ hc_eps,
                              const int* __restrict__ itersp, int iter) {
    if (iter >= itersp[0]) return;
    int t = threadIdx.x;
    if (t < E) {
        const float invE = 1.0f / (float)E;
        d1[t] = invE / (acc[t] + hc_eps[0]);
        acc[t] = 0.0f;
    }
}

// ---------------------------------------------------------------------------
// Kernel 4: out[t,e] = cost[t,e] * d0[t] * d1[e]  (in place on d_out).
__global__ void hc_scale_kernel(float* __restrict__ out,
                                const float* __restrict__ d0,
                                const float* __restrict__ d1,
                                int T) {
    const int lane = threadIdx.x & 31;
    const int wid  = (blockIdx.x * blockDim.x + threadIdx.x) >> 5;
    const int nw   = (gridDim.x * blockDim.x) >> 5;
    const v4f w = ((const v4f*)d1)[lane];
    for (int row = wid; row < T; row += nw) {
        float s = d0[row];
        v4f* pr = (v4f*)(out + (size_t)row * E);
        v4f c = pr[lane];
        c = c * (w * s);
        __builtin_nontemporal_store(c, &pr[lane]);   // output is write-once: NT
    }
}

// ---------------------------------------------------------------------------
extern "C" void kernel_launch(void* const* d_in, const int* in_sizes, int n_in,
                              void* d_out, int out_size, void* d_ws, size_t ws_size,
                              hipStream_t stream) {
    const float* x        = (const float*)d_in[0];
    const int*   hc_fn    = (const int*)  d_in[1];
    const float* hc_scale = (const float*)d_in[2];
    const float* hc_base  = (const float*)d_in[3];
    const float* hc_mult  = (const float*)d_in[4];
    const int*   iters    = (const int*)  d_in[5];
    const float* hc_eps   = (const float*)d_in[6];
    float* out = (float*)d_out;

    const int T = in_sizes[0] / E;

    // Workspace layout: d0[T] | d1[E] | acc[E]
    float* d0  = (float*)d_ws;
    float* d1  = d0 + T;
    float* acc = d1 + E;

    const int blocks = 2048, threads = 256;   // 16384 waves -> 16 rows per wave

    hc_init_kernel<<<1, E, 0, stream>>>(d1, acc);
    hc_act_kernel<<<blocks, threads, 0, stream>>>(x, out, d0, hc_fn, hc_scale,
                                                  hc_base, hc_mult, T);
    for (int i = 0; i < MAX_ITERS; ++i) {
        hc_pass_kernel<<<blocks, threads, 0, stream>>>(out, d0, d1, acc,
                                                       hc_eps, iters, i, T);
        hc_fin_kernel<<<1, E, 0, stream>>>(d1, acc, hc_eps, iters, i);
    }
    hc_scale_kernel<<<blocks, threads, 0, stream>>>(out, d0, d1, T);
}